// DimPositionalEmbedding_15676630631236
// MI455X (gfx1250) — compile-verified
//
#include <hip/hip_runtime.h>
#include <stdint.h>

// ---------------- problem constants (from reference) ----------------
#define NB   8        // batch
#define SEQ  4096     // sequence length
#define HID  1024     // embedding width
#define TPB  256      // threads per block (8 waves on wave32)
#define TOKS 8        // tokens handled per block in the gather kernel
// MAXL = {514, 256, 128}; init counters = {2, 0, 0}

typedef float v4f __attribute__((ext_vector_type(4)));

// ---------------- CDNA5 async-copy helpers (gfx1250 ISA) ----------------
// GLOBAL_LOAD_ASYNC_TO_LDS_B128: per-lane 16B copy global->LDS, tracked by
// ASYNCcnt (completes in order among async loads of the same wave).
__device__ __forceinline__ void async_load_b128_to_lds(unsigned lds_addr, uint64_t gaddr) {
  asm volatile("global_load_async_to_lds_b128 %0, %1, off"
               :: "v"(lds_addr), "v"(gaddr)
               : "memory");
}
__device__ __forceinline__ void wait_asynccnt_3() {
  asm volatile("s_wait_asynccnt 3" ::: "memory");
}
__device__ __forceinline__ void wait_asynccnt_0() {
  asm volatile("s_wait_asynccnt 0" ::: "memory");
}

// ---------------- kernel 1: per-sequence odometer scan ----------------
// One wave per sequence. Tokens staged through LDS in 512-token chunks.
// All 32 lanes run the (uniform) serial state machine; lane 0 writes the
// packed 25-bit map per token and the final counters.
__global__ __launch_bounds__(32)
void odometer_kernel(const int* __restrict__ ids,
                     unsigned* __restrict__ maps,     // [NB*SEQ] packed indices
                     float* __restrict__ cnt_out) {   // [NB*3] final counters
  const int b    = blockIdx.x;
  const int lane = threadIdx.x;
  const int* tok = ids + (size_t)b * SEQ;
  unsigned*  m   = maps + (size_t)b * SEQ;

  __shared__ int toks[512];

  int c0 = 2, c1 = 0, c2 = 0;       // init: [OFFSET, 0, 0]
  bool done = false;

  for (int base = 0; base < SEQ; base += 512) {
    for (int j = lane; j < 512; j += 32) toks[j] = tok[base + j];
    __syncthreads();
    for (int j = 0; j < 512; ++j) {
      const int t = toks[j];
      done = done || (t == 1);                       // EOS freezes state
      if (!done) {
        // KEY_TABLE: token 10 -> dim 1, token 11 -> dim 2, else dim 0
        const int c = (t == 10) ? 1 : ((t == 11) ? 2 : 0);
        // counters[c] += 1 ; counters[j<c] = 0 ; then carry-propagate
        int v0 = (c == 0) ? (c0 + 1) : 0;
        int v1 = (c == 1) ? (c1 + 1) : ((c == 2) ? 0 : c1);
        int v2 = (c == 2) ? (c2 + 1) : c2;
        int carry = (v0 >= 514); if (carry) v0 = 0;
        v1 += carry;
        carry = (v1 >= 256);     if (carry) v1 = 0;
        v2 += carry;
        if (v2 >= 128) v2 = 0;
        c0 = v0; c1 = v1; c2 = v2;
      }
      // after EOS the embedding map pins to MAXL-1
      const unsigned e0 = done ? 513u : (unsigned)c0;   // 10 bits
      const unsigned e1 = done ? 255u : (unsigned)c1;   //  8 bits
      const unsigned e2 = done ? 127u : (unsigned)c2;   //  7 bits
      if (lane == 0) m[base + j] = e0 | (e1 << 10) | (e2 << 18);
    }
    __syncthreads();
  }
  if (lane == 0) {
    cnt_out[b * 3 + 0] = (float)c0;
    cnt_out[b * 3 + 1] = (float)c1;
    cnt_out[b * 3 + 2] = (float)c2;
  }
}

// ---------------- kernel 2: async-pipelined gather + sum ----------------
// Each thread owns a 4-float column slice of a token's 1024-wide output row.
// Per token it issues 3 async b128 copies (emb0/1/2 row chunk -> its private
// LDS slot), double-buffered across tokens; waits via s_wait_asynccnt, sums
// from LDS, and NT-stores (output is streamed once; keep L2 for the tables).
__global__ __launch_bounds__(TPB)
void gather_sum_kernel(const float* __restrict__ e0,
                       const float* __restrict__ e1,
                       const float* __restrict__ e2,
                       const unsigned* __restrict__ maps,
                       float* __restrict__ out) {
  __shared__ v4f stage[2][TPB * 3];   // [buf][thread][table] : 24 KB
  const int    tid     = threadIdx.x;
  const size_t tokBase = (size_t)blockIdx.x * TOKS;
  const int    col     = tid * 4;     // float column within the 1024-wide row

  auto issue = [&](int buf, int i) {
    const unsigned p  = maps[tokBase + i];        // uniform per block-iteration
    const unsigned r0 = p & 1023u;
    const unsigned r1 = (p >> 10) & 255u;
    const unsigned r2 = (p >> 18) & 127u;
    v4f* slot = &stage[buf][tid * 3];
    async_load_b128_to_lds((unsigned)(uintptr_t)(slot + 0),
                           (uint64_t)(uintptr_t)(e0 + (size_t)r0 * HID + col));
    async_load_b128_to_lds((unsigned)(uintptr_t)(slot + 1),
                           (uint64_t)(uintptr_t)(e1 + (size_t)r1 * HID + col));
    async_load_b128_to_lds((unsigned)(uintptr_t)(slot + 2),
                           (uint64_t)(uintptr_t)(e2 + (size_t)r2 * HID + col));
  };

  issue(0, 0);
  for (int i = 0; i < TOKS; ++i) {
    if (i + 1 < TOKS) {
      issue((i + 1) & 1, i + 1);   // 6 outstanding: 3 (token i) + 3 (token i+1)
      wait_asynccnt_3();           // in-order completion -> token i's 3 are done
    } else {
      wait_asynccnt_0();
    }
    const v4f* slot = &stage[i & 1][tid * 3];
    const v4f a = slot[0];
    const v4f b = slot[1];
    const v4f c = slot[2];
    v4f r = a + b + c;
    __builtin_nontemporal_store(r, (v4f*)(out + (tokBase + i) * HID + col));
  }
}

// ---------------- host-side launch ----------------
extern "C" void kernel_launch(void* const* d_in, const int* in_sizes, int n_in,
                              void* d_out, int out_size, void* d_ws, size_t ws_size,
                              hipStream_t stream) {
  (void)in_sizes; (void)n_in; (void)out_size; (void)ws_size;
  const int*   ids = (const int*)  d_in[0];   // input_ids [8,4096] (JAX default int32)
  const float* e0  = (const float*)d_in[1];   // emb0 [514,1024]
  const float* e1  = (const float*)d_in[2];   // emb1 [256,1024]
  const float* e2  = (const float*)d_in[3];   // emb2 [128,1024]
  float* out = (float*)d_out;                 // pos [8,4096,1024] ++ final_counters [8,3]

  unsigned* maps = (unsigned*)d_ws;           // 8*4096*4 = 128 KB packed indices
  float* cnt_out = out + (size_t)NB * SEQ * HID;

  odometer_kernel<<<NB, 32, 0, stream>>>(ids, maps, cnt_out);
  gather_sum_kernel<<<(NB * SEQ) / TOKS, TPB, 0, stream>>>(e0, e1, e2, maps, out);
}